// GC_3D_v1_31568009625983
// MI455X (gfx1250) — compile-verified
//
#include <hip/hip_runtime.h>
#include <stdint.h>

#define USE_ASYNC_LDS 1

// Problem geometry (from reference: shape (2,1,128,192,192) f32)
constexpr int Bn = 2, Dd = 128, Hh = 192, Ww = 192;
constexpr int TZ = 8, TY = 8, TX = 32;           // tile per block (exactly divides dims)
constexpr int SZ = TZ + 1, SY = TY + 1, SX = TX + 1;
constexpr int TILE = SZ * SY * SX;               // 2673 halo elements
constexpr int NBX = Ww / TX;                     // 6
constexpr int NBY = Hh / TY;                     // 24
constexpr int NBZ = Dd / TZ;                     // 16
constexpr int NB  = NBX * NBY * NBZ * Bn;        // 4608 blocks
constexpr int NACC = 27;                         // 13 num + 13 den + 1 bce

typedef __attribute__((ext_vector_type(2))) float v2f;
typedef __attribute__((ext_vector_type(8))) float v8f;

// 13 neighbor directions; slicing => accesses only at base+{0,1} per dim
__device__ constexpr int DZv[13] = {1, 0, 0, 1, 1,  0, 0,  1,  1, 1,  1,  1, 1};
__device__ constexpr int DYv[13] = {0, 1, 0, 1, -1, 1, 1,  0,  0, -1, 1,  1, -1};
__device__ constexpr int DXv[13] = {0, 0, 1, 0, 0,  1, -1, -1, 1, -1, -1, 1, 1};

// Fused per-thread z-march. MASKED=false is the fast path for blocks that do not
// touch the far z/y/x faces (all edge masks provably true -> all selects fold away).
template <bool MASKED>
__device__ __forceinline__ void compute_tile(const float* __restrict__ in_s,
                                             const float* __restrict__ tg_s,
                                             const int lbase, const bool mx,
                                             const bool my, const int gz0,
                                             float acc[NACC]) {
  float ci[2][2][2], ct[2][2][2];
#pragma unroll
  for (int y = 0; y < 2; ++y)
#pragma unroll
    for (int x = 0; x < 2; ++x) {
      const int idx = lbase + y * SX + x;
      ci[0][y][x] = in_s[idx];
      ct[0][y][x] = tg_s[idx];
    }

  for (int lz = 0; lz < TZ; ++lz) {
    const int idx1 = lbase + (lz + 1) * (SY * SX);
#pragma unroll
    for (int y = 0; y < 2; ++y)
#pragma unroll
      for (int x = 0; x < 2; ++x) {
        ci[1][y][x] = in_s[idx1 + y * SX + x];
        ct[1][y][x] = tg_s[idx1 + y * SX + x];
      }
    const bool mz = MASKED ? ((gz0 + lz) < (Dd - 1)) : true;

    // BCE partial: t*max(log p,-100) + (1-t)*max(log1p(-p),-100)
    {
      const float p = ci[0][0][0];
      const float t = ct[0][0][0];
      acc[26] += t * fmaxf(logf(p), -100.0f) + (1.0f - t) * fmaxf(log1pf(-p), -100.0f);
    }

#pragma unroll
    for (int d = 0; d < 13; ++d) {
      const int dz = DZv[d], dy = DYv[d], dx = DXv[d];
      const int az = dz;
      const int ay = (dy == 1) ? 1 : 0;
      const int ax = (dx == 1) ? 1 : 0;
      const int byc = (dy == -1) ? 1 : 0;
      const int bxc = (dx == -1) ? 1 : 0;
      const float di = fabsf(ci[az][ay][ax] - ci[0][byc][bxc]);
      const float dt = fabsf(ct[az][ay][ax] - ct[0][byc][bxc]);
      if (MASKED) {
        bool v = true;
        if (dz != 0) v = v && mz;
        if (dy != 0) v = v && my;
        if (dx != 0) v = v && mx;
        acc[d]      += v ? di * dt : 0.0f;  // select (not multiply) kills halo garbage
        acc[13 + d] += v ? dt      : 0.0f;
      } else {
        acc[d]      += di * dt;
        acc[13 + d] += dt;
      }
    }

#pragma unroll
    for (int y = 0; y < 2; ++y)
#pragma unroll
      for (int x = 0; x < 2; ++x) {
        ci[0][y][x] = ci[1][y][x];
        ct[0][y][x] = ct[1][y][x];
      }
  }
}

__global__ __launch_bounds__(256)
void gc3d_partial(const float* __restrict__ inp, const float* __restrict__ tgt,
                  float* __restrict__ ws) {
  __shared__ float in_s[TILE];
  __shared__ float tg_s[TILE];
  __shared__ float S[256];   // [32 accumulators][8 waves]

  const int tid = threadIdx.x;
  const int bid = blockIdx.x;
  const int bx = bid % NBX;
  int tq = bid / NBX;
  const int by = tq % NBY; tq /= NBY;
  const int bz = tq % NBZ;
  const int bb = tq / NBZ;

  const int gx0 = bx * TX, gy0 = by * TY, gz0 = bz * TZ;
  const size_t batchOff = (size_t)bb * Dd * Hh * Ww;

  // ---- halo tile load: global -> LDS (CDNA5 async copy, bypasses VGPRs) ----
#if USE_ASYNC_LDS
  {
    const unsigned in_base = (unsigned)(uintptr_t)in_s;
    const unsigned tg_base = (unsigned)(uintptr_t)tg_s;
    for (int i = tid; i < TILE; i += 256) {
      const int x = i % SX;
      const int r = i / SX;
      const int y = r % SY;
      const int z = r / SY;
      int gz = gz0 + z; if (gz > Dd - 1) gz = Dd - 1;   // clamped halo; masked later
      int gy = gy0 + y; if (gy > Hh - 1) gy = Hh - 1;
      int gx = gx0 + x; if (gx > Ww - 1) gx = Ww - 1;
      const size_t g = batchOff + ((size_t)gz * Hh + gy) * Ww + gx;
      const unsigned long long pa = (unsigned long long)(uintptr_t)(inp + g);
      const unsigned long long pb = (unsigned long long)(uintptr_t)(tgt + g);
      const unsigned la = in_base + (unsigned)i * 4u;
      const unsigned lb = tg_base + (unsigned)i * 4u;
      asm volatile("global_load_async_to_lds_b32 %0, %1, off" :: "v"(la), "v"(pa) : "memory");
      asm volatile("global_load_async_to_lds_b32 %0, %1, off" :: "v"(lb), "v"(pb) : "memory");
    }
    asm volatile("s_wait_asynccnt 0" ::: "memory");
  }
#else
  for (int i = tid; i < TILE; i += 256) {
    const int x = i % SX;
    const int r = i / SX;
    const int y = r % SY;
    const int z = r / SY;
    int gz = gz0 + z; if (gz > Dd - 1) gz = Dd - 1;
    int gy = gy0 + y; if (gy > Hh - 1) gy = Hh - 1;
    int gx = gx0 + x; if (gx > Ww - 1) gx = Ww - 1;
    const size_t g = batchOff + ((size_t)gz * Hh + gy) * Ww + gx;
    in_s[i] = inp[g];
    tg_s[i] = tgt[g];
  }
#endif
  __syncthreads();

  const int tx = tid & 31, ty = tid >> 5;
  const int lbase = ty * SX + tx;
  const bool my = (gy0 + ty) < (Hh - 1);
  const bool mx = (gx0 + tx) < (Ww - 1);

  float acc[NACC];
#pragma unroll
  for (int a = 0; a < NACC; ++a) acc[a] = 0.0f;

  // Block-uniform specialization: ~75% of blocks never touch a far face.
  const bool interior = (bx < NBX - 1) && (by < NBY - 1) && (bz < NBZ - 1);
  if (interior)
    compute_tile<false>(in_s, tg_s, lbase, mx, my, gz0, acc);
  else
    compute_tile<true>(in_s, tg_s, lbase, mx, my, gz0, acc);

  // ---- intra-wave tree reduction (wave32) ----
#pragma unroll
  for (int a = 0; a < NACC; ++a) {
    float v = acc[a];
#pragma unroll
    for (int off = 16; off >= 1; off >>= 1) v += __shfl_xor(v, off, 32);
    acc[a] = v;
  }

  // ---- cross-wave reduction via WMMA: S is [32 acc][8 wave] ----
  S[tid] = 0.0f;
  __syncthreads();
  const int wave = tid >> 5, lane = tid & 31;
  if (lane == 0) {
#pragma unroll
    for (int a = 0; a < NACC; ++a) S[a * 8 + wave] = acc[a];
  }
  __syncthreads();

  if (wave < 2) {            // wave-uniform branch: EXEC stays all-ones for WMMA
    const int m0 = wave * 16;
    const int m  = m0 + (lane & 15);
    const int kb = (lane < 16) ? 0 : 2;   // 16x4 f32 A layout: v0={K0,K2}, v1={K1,K3}
    v8f c = {0.f, 0.f, 0.f, 0.f, 0.f, 0.f, 0.f, 0.f};
    v2f ones; ones.x = 1.0f; ones.y = 1.0f;
#pragma unroll
    for (int chunk = 0; chunk < 2; ++chunk) {   // K=4 per op, 8 waves = 2 chained ops
      v2f a;
      a.x = S[m * 8 + chunk * 4 + kb];
      a.y = S[m * 8 + chunk * 4 + kb + 1];
      // D = A(16x4 partials) x B(4x16 ones) + C  => exact f32 row sums
      c = __builtin_amdgcn_wmma_f32_16x16x4_f32(false, a, false, ones, (short)0, c,
                                                false, false);
    }
    float* Wp = ws + (size_t)bid * 32;
    // C/D layout: VGPR r holds M=r (lanes 0-15) and M=r+8 (lanes 16-31)
    if (lane == 0) {
#pragma unroll
      for (int r = 0; r < 8; ++r) Wp[m0 + r] = c[r];
    } else if (lane == 16) {
#pragma unroll
      for (int r = 0; r < 8; ++r) Wp[m0 + 8 + r] = c[r];
    }
  }
}

__global__ __launch_bounds__(256)
void gc3d_final(const float* __restrict__ Wp, float* __restrict__ out) {
  __shared__ double Rr[8][32];
  __shared__ double Tot[32];
  const int t = threadIdx.x;
  const int a = t & 31, chunk = t >> 5;
  constexpr int per = NB / 8;   // 576
  double s = 0.0;
  for (int j = chunk * per; j < (chunk + 1) * per; ++j)
    s += (double)Wp[(size_t)j * 32 + a];
  Rr[chunk][a] = s;
  __syncthreads();
  if (t < 32) {
    double tot = 0.0;
#pragma unroll
    for (int cch = 0; cch < 8; ++cch) tot += Rr[cch][t];
    Tot[t] = tot;
  }
  __syncthreads();
  if (t == 0) {
    double num = 0.0;
    for (int d = 0; d < 13; ++d) num += Tot[d] / (Tot[13 + d] + 1e-5);
    const double N = (double)Bn * Dd * Hh * Ww;
    const double bce = -Tot[26] / N;               // region term (LMDA = 1)
    out[0] = (float)(bce + (1.0 - num / 13.0));    // + boundary term
  }
}

extern "C" void kernel_launch(void* const* d_in, const int* in_sizes, int n_in,
                              void* d_out, int out_size, void* d_ws, size_t ws_size,
                              hipStream_t stream) {
  const float* inp = (const float*)d_in[0];
  const float* tgt = (const float*)d_in[1];
  float* out = (float*)d_out;
  float* ws  = (float*)d_ws;   // uses NB*32*4 = 576 KB, fully overwritten each call
  gc3d_partial<<<NB, 256, 0, stream>>>(inp, tgt, ws);
  gc3d_final<<<1, 256, 0, stream>>>(ws, out);
}